// GNN_43413529428621
// MI455X (gfx1250) — compile-verified
//
#include <hip/hip_runtime.h>
#include <hip/hip_bf16.h>
#include <math.h>

typedef float v2f __attribute__((ext_vector_type(2)));
typedef float v8f __attribute__((ext_vector_type(8)));
typedef int vint4 __attribute__((vector_size(16)));   // matches builtin param type

constexpr int NN = 2048;   // nodes
constexpr int BB = 8;      // batch
constexpr int DD = 6;      // input feature dim
constexpr int HH = 64;     // hidden dim

// ---------------------------------------------------------------------------
// gfx1250 async global->LDS copy (ASYNCcnt path), guarded by __has_builtin.
// Builtin signature (from hipcc diagnostic): (v4i AS1*, v4i AS3*, Ii, Ii).
// Falls back to a plain float4 copy (ds_store path) when unavailable.
// ---------------------------------------------------------------------------
#if defined(__has_builtin)
#if __has_builtin(__builtin_amdgcn_global_load_async_to_lds_b128)
#define HAS_ASYNC_LDS 1
#endif
#if __has_builtin(__builtin_amdgcn_s_wait_asynccnt)
#define HAS_WAIT_ASYNC_BUILTIN 1
#endif
#endif
#ifndef HAS_ASYNC_LDS
#define HAS_ASYNC_LDS 0
#endif
#ifndef HAS_WAIT_ASYNC_BUILTIN
#define HAS_WAIT_ASYNC_BUILTIN 0
#endif

__device__ __forceinline__ void lds_cp16(float* dst, const float* src) {
#if HAS_ASYNC_LDS
    float* s = const_cast<float*>(src);
    __builtin_amdgcn_global_load_async_to_lds_b128(
        (__attribute__((address_space(1))) vint4*)s,
        (__attribute__((address_space(3))) vint4*)dst,
        0, 0);
#else
    *reinterpret_cast<float4*>(dst) = *reinterpret_cast<const float4*>(src);
#endif
}

__device__ __forceinline__ void wait_async_all() {
#if HAS_ASYNC_LDS
#if HAS_WAIT_ASYNC_BUILTIN
    __builtin_amdgcn_s_wait_asynccnt(0);
#else
    asm volatile("s_wait_asynccnt 0" ::: "memory");
#endif
#endif
}

// ---------------------------------------------------------------------------
// adj_build: one workgroup per (row i, batch b).
// adj_ij = exp_ij * valid_i * valid_j / (sum_j exp_ij*valid_j + 1e-20*sum_j exp_ij)
// (the softmax denominator Z_i cancels between softmax and renormalization)
// ---------------------------------------------------------------------------
__launch_bounds__(256)
__global__ void adj_build(const float* __restrict__ emb,
                          const float* __restrict__ nb,
                          const float* __restrict__ sigma,
                          float* __restrict__ adj) {
    __shared__ float es[NN * 3];
    __shared__ float redZ[256];
    __shared__ float redM[256];
    __shared__ float sc_sh;
    const int b = blockIdx.y;
    const int i = blockIdx.x;
    const int tid = threadIdx.x;
    const float* eb = emb + (size_t)b * NN * DD;
    for (int idx = tid; idx < NN * 3; idx += 256) {
        int j = idx / 3, c = idx - j * 3;
        es[idx] = eb[(size_t)j * DD + c];
    }
    __syncthreads();
    const int nbc = (int)(nb[b] + 0.5f);
    const float sg = sigma[0];
    const float inv = 1.0f / (3.0f * sg * sg);   // mean over 3 spatial dims / sigma^2
    const float xi = es[i * 3], yi = es[i * 3 + 1], zi = es[i * 3 + 2];
    float Z = 0.f, M = 0.f;
    for (int j = tid; j < NN; j += 256) {
        float dx = xi - es[j * 3], dy = yi - es[j * 3 + 1], dz = zi - es[j * 3 + 2];
        float w = __expf(-(dx * dx + dy * dy + dz * dz) * inv);
        Z += w;
        if (j < nbc) M += w;
    }
    redZ[tid] = Z; redM[tid] = M;
    __syncthreads();
    for (int off = 128; off > 0; off >>= 1) {
        if (tid < off) { redZ[tid] += redZ[tid + off]; redM[tid] += redM[tid + off]; }
        __syncthreads();
    }
    if (tid == 0) {
        sc_sh = (i < nbc) ? 1.0f / (redM[0] + 1e-20f * redZ[0]) : 0.0f;
    }
    __syncthreads();
    const float scale = sc_sh;
    float* arow = adj + (size_t)b * NN * NN + (size_t)i * NN;
    for (int j = tid; j < NN; j += 256) {
        float dx = xi - es[j * 3], dy = yi - es[j * 3 + 1], dz = zi - es[j * 3 + 2];
        float w = __expf(-(dx * dx + dy * dy + dz * dz) * inv);
        arow[j] = (j < nbc) ? w * scale : 0.0f;
    }
}

// ---------------------------------------------------------------------------
// build_x0: masked emb padded from 6 to 16 columns (so layer-0 spread can use WMMA)
// ---------------------------------------------------------------------------
__global__ void build_x0(const float* __restrict__ emb,
                         const float* __restrict__ nb,
                         float* __restrict__ x0) {
    int idx = blockIdx.x * 256 + threadIdx.x;
    if (idx >= BB * NN * 16) return;
    int k = idx & 15;
    int node = idx >> 4;
    int b = node / NN;
    int i = node - b * NN;
    int nbc = (int)(nb[b] + 0.5f);
    float v = 0.f;
    if (k < DD && i < nbc) v = emb[(size_t)node * DD + k];
    x0[idx] = v;
}

// ---------------------------------------------------------------------------
// spread_wmma: out[b] = adj[b] (NNxNN) @ x[b] (NNxF) via v_wmma_f32_16x16x4_f32.
// Double-buffered LDS tiles; tile t+1 is copied (async global->LDS when
// available) while WMMAs consume tile t.
// WMMA f32 operand mapping (wave32):
//   A 16x4 : lane l holds A[l&15][k0], A[l&15][k0+1], k0=(l>>4)*2  (2 VGPRs)
//   B 4x16 : lane l holds B[k0][l&15], B[k0+1][l&15]
//   C 16x16: vgpr v, lane l -> C[v + (l>>4)*8][l&15]
// ---------------------------------------------------------------------------
template <int F, int MT, int KT>
__launch_bounds__(256)
__global__ void spread_wmma(const float* __restrict__ adj,
                            const float* __restrict__ x,
                            float* __restrict__ out) {
    constexpr int NSUB = (MT / 16) * (F / 16) / 8;
    static_assert(NSUB >= 1, "tile too small");
    __shared__ __align__(16) float la[2][MT * KT];
    __shared__ __align__(16) float lb[2][KT * F];
    const int b = blockIdx.y;
    const int m0 = blockIdx.x * MT;
    const float* __restrict__ adjb = adj + (size_t)b * NN * NN + (size_t)m0 * NN;
    const float* __restrict__ xb = x + (size_t)b * NN * F;
    const int tid = threadIdx.x;
    const int lane = tid & 31;
    const int wave = tid >> 5;
    const int lrow = lane & 15;
    const int kb = (lane >> 4) * 2;

    v8f acc[NSUB];
#pragma unroll
    for (int s = 0; s < NSUB; s++)
#pragma unroll
        for (int q = 0; q < 8; q++) acc[s][q] = 0.0f;

    auto load_tile = [&](int k0, int buf) {
        // A tile: MT x KT, 16B chunks
        for (int idx = tid; idx < MT * KT / 4; idx += 256) {
            int r = idx / (KT / 4);
            int c4 = (idx - r * (KT / 4)) * 4;
            lds_cp16(&la[buf][r * KT + c4], &adjb[(size_t)r * NN + k0 + c4]);
        }
        // B tile: KT x F, 16B chunks
        for (int idx = tid; idx < KT * F / 4; idx += 256) {
            int r = idx / (F / 4);
            int c4 = (idx - r * (F / 4)) * 4;
            lds_cp16(&lb[buf][r * F + c4], &xb[(size_t)(k0 + r) * F + c4]);
        }
    };

    constexpr int NTILES = NN / KT;
    load_tile(0, 0);
    for (int t = 0; t < NTILES; t++) {
        wait_async_all();
        __syncthreads();
        if (t + 1 < NTILES) load_tile((t + 1) * KT, (t + 1) & 1);
        const float* __restrict__ A = la[t & 1];
        const float* __restrict__ Bt = lb[t & 1];
#pragma unroll 4
        for (int kk = 0; kk < KT; kk += 4) {
#pragma unroll
            for (int s = 0; s < NSUB; s++) {
                const int st = wave * NSUB + s;
                const int mi = st / (F / 16);
                const int ni = st % (F / 16);
                v2f a, bf;
                a.x = A[(mi * 16 + lrow) * KT + kk + kb];
                a.y = A[(mi * 16 + lrow) * KT + kk + kb + 1];
                bf.x = Bt[(kk + kb) * F + ni * 16 + lrow];
                bf.y = Bt[(kk + kb + 1) * F + ni * 16 + lrow];
                acc[s] = __builtin_amdgcn_wmma_f32_16x16x4_f32(
                    false, a, false, bf, (short)0, acc[s], false, false);
            }
        }
    }

    float* __restrict__ ob = out + (size_t)b * NN * F + (size_t)m0 * F;
#pragma unroll
    for (int s = 0; s < NSUB; s++) {
        const int st = wave * NSUB + s;
        const int mi = st / (F / 16);
        const int ni = st % (F / 16);
        const int col = ni * 16 + lrow;
        const int rbase = mi * 16 + (lane >> 4) * 8;
#pragma unroll
        for (int v = 0; v < 8; v++)
            ob[(size_t)(rbase + v) * F + col] = acc[s][v];
    }
}

// ---------------------------------------------------------------------------
// dense_wmma: e_out = act( concat(spread, x) @ [WA|WB] + [bA|bB] )
// rows = B*NN flattened; K = 2*KHALF (12 for layer-0, 128 for layers 1..3);
// ReLU on the first 32 output channels. f32 WMMA, K stepped by 4.
// ---------------------------------------------------------------------------
template <int KHALF, int SSTR>
__launch_bounds__(256)
__global__ void dense_wmma(const float* __restrict__ spr,
                           const float* __restrict__ xs,
                           const float* __restrict__ WAp,
                           const float* __restrict__ WBp,
                           const float* __restrict__ bAp,
                           const float* __restrict__ bBp,
                           float* __restrict__ eout) {
    constexpr int K = 2 * KHALF;
    constexpr int MT = 64;
    __shared__ __align__(16) float la[MT * K];
    __shared__ __align__(16) float lw[K * HH];
    const int tid = threadIdx.x;
    const int m0 = blockIdx.x * MT;  // global node index base

    if constexpr ((KHALF % 4) == 0) {
        // 16B-chunk staging (async path when available)
        for (int idx = tid; idx < K * HH / 4; idx += 256) {
            int k = idx >> 4;
            int f4 = (idx & 15) * 4;
            const float* src = (f4 < 32) ? (WAp + k * 32 + f4) : (WBp + k * 32 + (f4 - 32));
            lds_cp16(&lw[k * HH + f4], src);
        }
        for (int idx = tid; idx < MT * K / 4; idx += 256) {
            int r = idx / (K / 4);
            int c4 = (idx - r * (K / 4)) * 4;
            size_t row = (size_t)(m0 + r);
            const float* src = (c4 < KHALF) ? (spr + row * SSTR + c4)
                                            : (xs + row * SSTR + (c4 - KHALF));
            lds_cp16(&la[r * K + c4], src);
        }
        wait_async_all();
    } else {
        for (int idx = tid; idx < K * HH; idx += 256) {
            int k = idx >> 6, f = idx & 63;
            lw[idx] = (f < 32) ? WAp[k * 32 + f] : WBp[k * 32 + (f - 32)];
        }
        for (int idx = tid; idx < MT * K; idx += 256) {
            int r = idx / K, c = idx - r * K;
            size_t row = (size_t)(m0 + r);
            la[idx] = (c < KHALF) ? spr[row * SSTR + c] : xs[row * SSTR + (c - KHALF)];
        }
    }
    __syncthreads();

    const int lane = tid & 31;
    const int wave = tid >> 5;
    const int lrow = lane & 15;
    const int kb = (lane >> 4) * 2;
    v8f acc[2];
#pragma unroll
    for (int s = 0; s < 2; s++)
#pragma unroll
        for (int q = 0; q < 8; q++) acc[s][q] = 0.0f;

#pragma unroll 4
    for (int kk = 0; kk < K; kk += 4) {
#pragma unroll
        for (int s = 0; s < 2; s++) {
            const int st = wave * 2 + s;
            const int mi = st >> 2;
            const int ni = st & 3;
            v2f a, bf;
            a.x = la[(mi * 16 + lrow) * K + kk + kb];
            a.y = la[(mi * 16 + lrow) * K + kk + kb + 1];
            bf.x = lw[(kk + kb) * HH + ni * 16 + lrow];
            bf.y = lw[(kk + kb + 1) * HH + ni * 16 + lrow];
            acc[s] = __builtin_amdgcn_wmma_f32_16x16x4_f32(
                false, a, false, bf, (short)0, acc[s], false, false);
        }
    }

#pragma unroll
    for (int s = 0; s < 2; s++) {
        const int st = wave * 2 + s;
        const int mi = st >> 2;
        const int ni = st & 3;
        const int col = ni * 16 + lrow;
        const float bias = (col < 32) ? bAp[col] : bBp[col - 32];
        const int rbase = mi * 16 + (lane >> 4) * 8;
#pragma unroll
        for (int v = 0; v < 8; v++) {
            float val = acc[s][v] + bias;
            if (col < 32) val = fmaxf(val, 0.0f);  // ReLU on A half only
            eout[(size_t)(m0 + rbase + v) * HH + col] = val;
        }
    }
}

// ---------------------------------------------------------------------------
// stats_k: masked per-feature mean + 1/sqrt(var) over valid nodes. One WG per b.
// ---------------------------------------------------------------------------
__launch_bounds__(256)
__global__ void stats_k(const float* __restrict__ e,
                        const float* __restrict__ nb,
                        float* __restrict__ st) {
    __shared__ float s1[256];
    __shared__ float s2[256];
    const int b = blockIdx.x;
    const int tid = threadIdx.x;
    const int f = tid & 63, g = tid >> 6;
    const int nbc = (int)(nb[b] + 0.5f);
    const float* eb = e + (size_t)b * NN * HH;
    float a1 = 0.f, a2 = 0.f;
    for (int i = g; i < nbc; i += 4) {
        float v = eb[(size_t)i * HH + f];
        a1 += v; a2 += v * v;
    }
    s1[tid] = a1; s2[tid] = a2;
    __syncthreads();
    if (tid < 64) {
        float t1 = s1[tid] + s1[tid + 64] + s1[tid + 128] + s1[tid + 192];
        float t2 = s2[tid] + s2[tid + 64] + s2[tid + 128] + s2[tid + 192];
        float nf = nb[b];
        float mean = t1 / (nf + 1e-20f);
        float var = (t2 - 2.f * mean * t1 + nf * mean * mean) / (nf + 1e-20f) + 1e-20f;
        st[b * 128 + tid] = mean;
        st[b * 128 + 64 + tid] = rsqrtf(var);
    }
}

// ---------------------------------------------------------------------------
// norm_k: xn = valid_i * (e - mean) * rstd
// ---------------------------------------------------------------------------
__global__ void norm_k(const float* __restrict__ e,
                       const float* __restrict__ st,
                       const float* __restrict__ nb,
                       float* __restrict__ xn) {
    int idx = blockIdx.x * 256 + threadIdx.x;
    if (idx >= BB * NN * HH) return;
    int f = idx & 63;
    int node = idx >> 6;
    int b = node / NN;
    int i = node - b * NN;
    int nbc = (int)(nb[b] + 0.5f);
    float v = 0.f;
    if (i < nbc) v = (e[idx] - st[b * 128 + f]) * st[b * 128 + 64 + f];
    xn[idx] = v;
}

// ---------------------------------------------------------------------------
// readout_k: s = masked node-sum, layer-norm over features, sigmoid(s @ Wro + bro)
// ---------------------------------------------------------------------------
__launch_bounds__(256)
__global__ void readout_k(const float* __restrict__ e,
                          const float* __restrict__ nb,
                          const float* __restrict__ Wro,
                          const float* __restrict__ bro,
                          float* __restrict__ out) {
    __shared__ float s1[256];
    __shared__ float sf[64];
    const int b = blockIdx.x;
    const int tid = threadIdx.x;
    const int f = tid & 63, g = tid >> 6;
    const int nbc = (int)(nb[b] + 0.5f);
    const float* eb = e + (size_t)b * NN * HH;
    float a = 0.f;
    for (int i = g; i < nbc; i += 4) a += eb[(size_t)i * HH + f];
    s1[tid] = a;
    __syncthreads();
    if (tid < 64) sf[tid] = s1[tid] + s1[tid + 64] + s1[tid + 128] + s1[tid + 192];
    __syncthreads();
    if (tid == 0) {
        float m = 0.f;
        for (int q = 0; q < 64; q++) m += sf[q];
        m *= (1.0f / 64.0f);
        float v = 0.f;
        for (int q = 0; q < 64; q++) { float d = sf[q] - m; v += d * d; }
        v *= (1.0f / 64.0f);
        float rs = rsqrtf(v + 1e-5f);
        float dot = 0.f;
        for (int q = 0; q < 64; q++) dot += (sf[q] - m) * rs * Wro[q];
        dot += bro[0];
        out[b] = 1.0f / (1.0f + __expf(-dot));
    }
}

// ---------------------------------------------------------------------------
extern "C" void kernel_launch(void* const* d_in, const int* in_sizes, int n_in,
                              void* d_out, int out_size, void* d_ws, size_t ws_size,
                              hipStream_t stream) {
    const float* emb   = (const float*)d_in[0];
    // d_in[1] = mask (B,N,N) -- never read; reconstructed from batch_nb_nodes
    const float* nb    = (const float*)d_in[2];
    const float* sigma = (const float*)d_in[3];
    const float* W0A   = (const float*)d_in[4];
    const float* b0A   = (const float*)d_in[5];
    const float* W0B   = (const float*)d_in[6];
    const float* b0B   = (const float*)d_in[7];
    const float* WA    = (const float*)d_in[8];
    const float* bA    = (const float*)d_in[9];
    const float* WB    = (const float*)d_in[10];
    const float* bB    = (const float*)d_in[11];
    const float* Wro   = (const float*)d_in[12];
    const float* bro   = (const float*)d_in[13];
    float* out = (float*)d_out;

    char* ws = (char*)d_ws;
    size_t off = 0;
    float* adj  = (float*)(ws + off); off += (size_t)BB * NN * NN * 4;  // 134 MB
    float* xbuf = (float*)(ws + off); off += (size_t)BB * NN * HH * 4;  // 4 MB
    float* sbuf = (float*)(ws + off); off += (size_t)BB * NN * HH * 4;  // 4 MB
    float* ebuf = (float*)(ws + off); off += (size_t)BB * NN * HH * 4;  // 4 MB
    float* stats = (float*)(ws + off);                                   // 4 KB

    // 1) adjacency (folded double-softmax normalization)
    adj_build<<<dim3(NN, BB), 256, 0, stream>>>(emb, nb, sigma, adj);

    // 2) layer 0: pad emb to 16 cols, spread via WMMA, dense K=12
    build_x0<<<(BB * NN * 16 + 255) / 256, 256, 0, stream>>>(emb, nb, xbuf);
    spread_wmma<16, 128, 64><<<dim3(NN / 128, BB), 256, 0, stream>>>(adj, xbuf, sbuf);
    dense_wmma<DD, 16><<<BB * NN / 64, 256, 0, stream>>>(sbuf, xbuf, W0A, W0B, b0A, b0B, ebuf);

    // 3) layers 1..3: norm -> spread -> dense
    for (int l = 0; l < 3; l++) {
        stats_k<<<BB, 256, 0, stream>>>(ebuf, nb, stats);
        norm_k<<<(BB * NN * HH + 255) / 256, 256, 0, stream>>>(ebuf, stats, nb, xbuf);
        spread_wmma<HH, 64, 64><<<dim3(NN / 64, BB), 256, 0, stream>>>(adj, xbuf, sbuf);
        dense_wmma<HH, HH><<<BB * NN / 64, 256, 0, stream>>>(
            sbuf, xbuf,
            WA + (size_t)l * 2 * HH * 32, WB + (size_t)l * 2 * HH * 32,
            bA + (size_t)l * 32, bB + (size_t)l * 32, ebuf);
    }

    // 4) readout
    readout_k<<<BB, 256, 0, stream>>>(ebuf, nb, Wro, bro, out);
}